// differentiableSuperpixelEmbedding_38946763440627
// MI455X (gfx1250) — compile-verified
//
#include <hip/hip_runtime.h>

// Problem constants (from reference)
#define BATCH 8
#define CHN   3
#define HW    224
#define SEG   196
#define DIM   768
#define PP    14          // patches per side
#define FIN   588         // CHN * PP * PP
#define HWSZ  (HW*HW)     // 50176
#define ROWS  (BATCH*SEG) // 1568

typedef float v2f __attribute__((ext_vector_type(2)));
typedef float v8f __attribute__((ext_vector_type(8)));

// ---------------------------------------------------------------- zero init
__global__ void zero_f32_kernel(float* __restrict__ p, long long n) {
  long long i = (long long)blockIdx.x * blockDim.x + threadIdx.x;
  long long stride = (long long)gridDim.x * blockDim.x;
  for (; i < n; i += stride) p[i] = 0.0f;
}

__global__ void zero_i32_kernel(int* __restrict__ p, int n) {
  int i = blockIdx.x * blockDim.x + threadIdx.x;
  if (i < n) p[i] = 0;
}

// ------------------------------------------------- argmax + patch pooling
// One block = one (b, p, q) 16x16 patch. 256 threads = 256 pixels.
__global__ __launch_bounds__(256) void patch_pool_kernel(
    const float* __restrict__ img,     // [B,C,H,W]
    const float* __restrict__ seg,     // [B,S,H,W]
    float* __restrict__ pooled,        // [B,S,588]  (pre-divided by 256)
    int* __restrict__ count)           // [B,S]
{
  __shared__ int   lds_label[256];
  __shared__ float lds_img[3][256];

  const int blk = blockIdx.x;                 // b*196 + p*14 + q
  const int b   = blk / (PP*PP);
  const int pq  = blk % (PP*PP);
  const int p   = pq / PP;
  const int q   = pq % PP;
  const int tid = threadIdx.x;
  const int ky  = tid >> 4;
  const int kx  = tid & 15;
  const int h   = p * 16 + ky;
  const int w   = q * 16 + kx;

  const size_t pix = (size_t)h * HW + w;

  // --- per-pixel argmax over 196 segment logits (first-max tie rule) ---
  const float* sp = seg + (size_t)b * SEG * HWSZ + pix;
  float best = sp[0];
  int   bi   = 0;
  #pragma unroll 4
  for (int s = 1; s < SEG; ++s) {
    float v = sp[(size_t)s * HWSZ];
    if (s + 8 < SEG)
      __builtin_prefetch(sp + (size_t)(s + 8) * HWSZ, 0, 1); // global_prefetch_b8
    if (v > best) { best = v; bi = s; }
  }
  lds_label[tid] = bi;

  const float* ip = img + (size_t)b * CHN * HWSZ + pix;
  lds_img[0][tid] = ip[0];
  lds_img[1][tid] = ip[HWSZ];
  lds_img[2][tid] = ip[2 * HWSZ];
  __syncthreads();

  // --- deterministic fixed-order reduction: thread s owns segment s ---
  if (tid < SEG) {
    float s0 = 0.f, s1 = 0.f, s2 = 0.f;
    int cnt = 0;
    #pragma unroll 8
    for (int i = 0; i < 256; ++i) {       // LDS broadcast reads (same addr all lanes)
      if (lds_label[i] == tid) {
        s0 += lds_img[0][i];
        s1 += lds_img[1][i];
        s2 += lds_img[2][i];
        ++cnt;
      }
    }
    const float inv = 1.0f / 256.0f;
    float* pr = pooled + (size_t)(b * SEG + tid) * FIN + p * PP + q;
    pr[0]           = s0 * inv;           // c = 0  (k = 0*196 + p*14 + q)
    pr[PP * PP]     = s1 * inv;           // c = 1
    pr[2 * PP * PP] = s2 * inv;           // c = 2
    if (cnt) atomicAdd(&count[b * SEG + tid], cnt); // integer: deterministic
  }
}

// ------------------------------------------------- compaction position map
__global__ void build_posmap_kernel(const int* __restrict__ count,
                                    int* __restrict__ posmap) {
  const int b = blockIdx.x;
  if (threadIdx.x == 0) {
    int pos = 0;
    for (int s = 0; s < SEG; ++s) {
      int c = count[b * SEG + s];
      posmap[b * SEG + s] = (c > 0) ? pos : -1;
      if (c > 0) ++pos;
    }
  }
}

// ------------------------------------------------- WMMA f32 GEMM + scatter
// feat[row, :] = pooled[row, :] @ w_proj + b_proj  ; out[b, posmap[row]] = feat
// M = 1568 (98 tiles), N = 768 (48 tiles), K = 588 (147 x 4)
// Each wave: 1 M-tile x 4 N-tiles. 1176 waves = 147 blocks x 8 waves.
__global__ __launch_bounds__(256) void gemm_wmma_kernel(
    const float* __restrict__ A,       // [1568, 588]
    const float* __restrict__ Wp,      // [588, 768]
    const float* __restrict__ bias,    // [768]
    const int*   __restrict__ posmap,  // [1568]
    float* __restrict__ out)           // [B, S, 768] (pre-zeroed)
{
  const int lane = threadIdx.x & 31;
  const int wave = threadIdx.x >> 5;
  const int wg   = blockIdx.x * 8 + wave;   // 0..1175
  const int mtile  = wg / 12;               // 0..97
  const int ngroup = wg % 12;               // 0..11  -> 4 N-tiles each

  const int half = lane >> 4;               // 0: lanes 0-15, 1: lanes 16-31
  const int l16  = lane & 15;

  const int row0  = mtile * 16;
  const int ncol0 = ngroup * 64 + l16;      // column this lane owns in tile 0

  // A fragment: lane row = l16; VGPR0 holds K = 2*half, VGPR1 holds K = 2*half+1
  const float* Arow = A + (size_t)(row0 + l16) * FIN + 2 * half;
  // B fragment base: same K convention, column = ncol0 (+16/+32/+48 per tile)
  const float* Wc0 = Wp + (size_t)(2 * half) * DIM + ncol0;

  v8f acc0 = {}, acc1 = {}, acc2 = {}, acc3 = {};

  for (int k = 0; k < FIN; k += 4) {
    v2f a = *(const v2f*)(Arow + k);                 // 8B aligned
    const float* wk = Wc0 + (size_t)k * DIM;
    v2f b0 = { wk[0],       wk[DIM]      };
    v2f b1 = { wk[16],      wk[DIM + 16] };
    v2f b2 = { wk[32],      wk[DIM + 32] };
    v2f b3 = { wk[48],      wk[DIM + 48] };
    acc0 = __builtin_amdgcn_wmma_f32_16x16x4_f32(false, a, false, b0, (short)0, acc0, false, false);
    acc1 = __builtin_amdgcn_wmma_f32_16x16x4_f32(false, a, false, b1, (short)0, acc1, false, false);
    acc2 = __builtin_amdgcn_wmma_f32_16x16x4_f32(false, a, false, b2, (short)0, acc2, false, false);
    acc3 = __builtin_amdgcn_wmma_f32_16x16x4_f32(false, a, false, b3, (short)0, acc3, false, false);
  }

  const float bv0 = bias[ncol0];
  const float bv1 = bias[ncol0 + 16];
  const float bv2 = bias[ncol0 + 32];
  const float bv3 = bias[ncol0 + 48];

  // D layout: VGPR r -> row (r + 8*half), col = l16 (+tile offset)
  #pragma unroll
  for (int r = 0; r < 8; ++r) {
    const int row  = row0 + r + 8 * half;
    const int dest = posmap[row];
    if (dest >= 0) {
      const int bb = row / SEG;
      float* o = out + (size_t)(bb * SEG + dest) * DIM;
      o[ncol0]      = acc0[r] + bv0;
      o[ncol0 + 16] = acc1[r] + bv1;
      o[ncol0 + 32] = acc2[r] + bv2;
      o[ncol0 + 48] = acc3[r] + bv3;
    }
  }
}

// ---------------------------------------------------------------- launcher
extern "C" void kernel_launch(void* const* d_in, const int* in_sizes, int n_in,
                              void* d_out, int out_size, void* d_ws, size_t ws_size,
                              hipStream_t stream) {
  const float* img = (const float*)d_in[0];   // [8,3,224,224]
  const float* seg = (const float*)d_in[1];   // [8,196,224,224]
  const float* wpj = (const float*)d_in[2];   // [588,768]
  const float* bpj = (const float*)d_in[3];   // [768]
  float* out = (float*)d_out;                 // [8,196,768]

  // workspace layout
  char* ws = (char*)d_ws;
  float* pooled = (float*)ws;                             // 1568*588*4 = 3,687,936 B
  int*   count  = (int*)(ws + (size_t)ROWS * FIN * 4);    // 1568*4 B
  int*   posmap = (int*)(ws + (size_t)ROWS * FIN * 4 + 8192);

  // must re-init every call (out/ws are poisoned, not restored between replays)
  zero_f32_kernel<<<2048, 256, 0, stream>>>(out, (long long)out_size);
  zero_i32_kernel<<<(ROWS + 255) / 256, 256, 0, stream>>>(count, ROWS);

  patch_pool_kernel<<<BATCH * PP * PP, 256, 0, stream>>>(img, seg, pooled, count);
  build_posmap_kernel<<<BATCH, 32, 0, stream>>>(count, posmap);
  gemm_wmma_kernel<<<147, 256, 0, stream>>>(pooled, wpj, bpj, posmap, out);
}